// FNO3d_6880537608938
// MI455X (gfx1250) — compile-verified
//
#include <hip/hip_runtime.h>
#include <math.h>

// ---------------------------------------------------------------------------
// FNO3d forward for MI455X (gfx1250, wave32).
// f32 end-to-end (memory-bound workload; 23.3 TB/s HBM dominates).
// Channel matmuls use V_WMMA_F32_16X16X4_F32.
// Spectral conv = partial DFT matmuls over kept modes only (no full FFT).
// ---------------------------------------------------------------------------

typedef __attribute__((ext_vector_type(2))) float v2f;
typedef __attribute__((ext_vector_type(8))) float v8f;

#define NXD 64
#define NYD 64
#define NZP 46          // padded z (40 + 6)
#define NZ0 40
#define CHN 32
#define NB  4
#define SPAT (NXD*NYD*NZP)          // 188416
#define FIELD (NB*CHN*SPAT)         // 24117248 floats
#define KK 12                       // kept z modes
#define XM 24                       // kept x (and y) modes: 12 low + 12 high
#define NBE (NB*CHN*NXD*XM*KK)      // 2359296 complex
#define NC  (NB*CHN*XM*XM*KK)       // 884736 complex

#if defined(__has_builtin)
#if __has_builtin(__builtin_amdgcn_wmma_f32_16x16x4_f32)
#define HAVE_WMMA_F32 1
#endif
#endif

// D = A(16x4) x B(4x16) + C, f32. A: lane<16 -> M=lane, K={0,1}; lane>=16 ->
// M=lane-16, K={2,3}. B mirrors the K split. C/D: lane holds col N=lane&15,
// rows M = v + 8*(lane>>4).
__device__ __forceinline__ v8f wmma4(v2f a, v2f b, v8f c) {
#if defined(HAVE_WMMA_F32)
  return __builtin_amdgcn_wmma_f32_16x16x4_f32(false, a, false, b, (short)0, c,
                                               false, false);
#else
  int lane = threadIdx.x & 31;
  int half = lane >> 4;
  int N = lane & 15;
#pragma unroll
  for (int k = 0; k < 4; ++k) {
    float bk = __shfl((k & 1) ? b.y : b.x, N + 16 * (k >> 1), 32);
#pragma unroll
    for (int v = 0; v < 8; ++v) {
      int M = v + 8 * half;
      float aMk = __shfl((k & 1) ? a.y : a.x, M + 16 * (k >> 1), 32);
      c[v] += aMk * bk;
    }
  }
  return c;
#endif
}

__device__ __forceinline__ float gelu_f(float v) {
  return 0.5f * v * (1.0f + erff(v * 0.70710678118654752440f));
}

// out[nt] = bias + T(16pts x 32ch) x W(32x32, LDS layout [i][o] stride 33)
__device__ __forceinline__ void gemm16x32(const float* T, const float* Wm,
                                          const float* bias, int lane,
                                          v8f out[2]) {
  int M = lane & 15, half = lane >> 4;
#pragma unroll
  for (int nt = 0; nt < 2; ++nt) {
    float bv = bias[nt * 16 + M];
    v8f c;
#pragma unroll
    for (int v = 0; v < 8; ++v) c[v] = bv;
#pragma unroll
    for (int k0 = 0; k0 < 32; k0 += 4) {
      v2f a, bb;
      a.x = T[M * 33 + k0 + 2 * half];
      a.y = T[M * 33 + k0 + 2 * half + 1];
      int N = nt * 16 + M;
      bb.x = Wm[(k0 + 2 * half) * 33 + N];
      bb.y = Wm[(k0 + 2 * half + 1) * 33 + N];
      c = wmma4(a, bb, c);
    }
    out[nt] = c;
  }
}

// --------------------------- twiddle tables --------------------------------
__global__ void k_tables(float* zc, float* zs, float* yc, float* ys) {
  int t = blockIdx.x * blockDim.x + threadIdx.x;
  int stride = gridDim.x * blockDim.x;
  for (int idx = t; idx < 46 * 12; idx += stride) {
    int n = idx / 12, k = idx % 12;
    double a = 6.2831853071795864769 * (double)(n * k) / 46.0;
    zc[idx] = (float)cos(a);
    zs[idx] = (float)sin(a);
  }
  for (int idx = t; idx < 64 * 24; idx += stride) {
    int x = idx / 24, m = idx % 24;
    int mm = (m < 12) ? m : m + 40;  // high modes 52..63
    double a = 6.2831853071795864769 * (double)(x * mm) / 64.0;
    yc[idx] = (float)cos(a);
    ys[idx] = (float)sin(a);
  }
}

// --------------------------- lift: concat grid + 13->32 --------------------
__global__ void __launch_bounds__(256) k_lift(const float* __restrict__ xin,
                                              const float* __restrict__ pw,
                                              const float* __restrict__ pb,
                                              float* __restrict__ field) {
  __shared__ float sPw[13 * 32];
  __shared__ float sPb[32];
  int t = threadIdx.x;
  for (int i = t; i < 416; i += 256) sPw[i] = pw[i];
  if (t < 32) sPb[t] = pb[t];
  __syncthreads();
  int tid = blockIdx.x * 256 + t;
  int b = tid / SPAT;
  int q = tid % SPAT;
  int z = q % 46, y = (q / 46) % 64, xx = q / 2944;
  if (z < NZ0) {
    float acc[32];
#pragma unroll
    for (int o = 0; o < 32; ++o) acc[o] = sPb[o];
    const float* xp = xin + ((((b * 64 + xx) * 64 + y) * 40) + z) * 10;
    float vin[13];
#pragma unroll
    for (int ci = 0; ci < 10; ++ci) vin[ci] = xp[ci];
    vin[10] = (float)xx * (1.0f / 63.0f);
    vin[11] = (float)y * (1.0f / 63.0f);
    vin[12] = (float)z * (1.0f / 39.0f);
#pragma unroll
    for (int ci = 0; ci < 13; ++ci) {
      float v = vin[ci];
#pragma unroll
      for (int o = 0; o < 32; ++o) acc[o] += v * sPw[ci * 32 + o];
    }
#pragma unroll
    for (int o = 0; o < 32; ++o) field[(b * 32 + o) * SPAT + q] = acc[o];
  } else {
#pragma unroll
    for (int o = 0; o < 32; ++o) field[(b * 32 + o) * SPAT + q] = 0.0f;
  }
}

// ------------------- forward partial DFT: z then y, fused ------------------
// one block per (b,c,x) plane
__global__ void __launch_bounds__(384)
k_fwd_zy(const float* __restrict__ field, const float* __restrict__ zc,
         const float* __restrict__ zs, const float* __restrict__ yc,
         const float* __restrict__ ys, float* __restrict__ Bre,
         float* __restrict__ Bim) {
  __shared__ float ft[2944];
  __shared__ float Ar[768], Ai[768];
  int t = threadIdx.x;
  int x = blockIdx.x & 63, c = (blockIdx.x >> 6) & 31, b = blockIdx.x >> 11;
  const float* fp = field + ((b * 32 + c) * 64 + x) * 2944;
  for (int idx = t; idx < 2944; idx += 384) ft[idx] = fp[idx];
  __syncthreads();
  for (int idx = t; idx < 768; idx += 384) {  // A[y][k]: rfft along z
    int y = idx / 12, k = idx % 12;
    float sr = 0.f, si = 0.f;
    for (int n = 0; n < 46; ++n) {
      float f = ft[y * 46 + n];
      sr += f * zc[n * 12 + k];
      si -= f * zs[n * 12 + k];
    }
    Ar[idx] = sr;
    Ai[idx] = si;
  }
  __syncthreads();
  if (t < 288) {  // B[ym][k]: fft along y, kept modes only
    int ym = t / 12, k = t % 12;
    float br = 0.f, bi = 0.f;
    for (int y = 0; y < 64; ++y) {
      float ar = Ar[y * 12 + k], ai = Ai[y * 12 + k];
      float cy = yc[y * 24 + ym], sy = ys[y * 24 + ym];
      br += ar * cy + ai * sy;
      bi += ai * cy - ar * sy;
    }
    int ob = ((b * 32 + c) * 64 + x) * 288 + t;
    Bre[ob] = br;
    Bim[ob] = bi;
  }
}

// forward partial DFT along x; one block per (b,c,xm), coalesced over (ym,k)
__global__ void __launch_bounds__(288)
k_fwd_x(const float* __restrict__ Bre, const float* __restrict__ Bim,
        const float* __restrict__ yc, const float* __restrict__ ys,
        float* __restrict__ Cre, float* __restrict__ Cim) {
  int t = threadIdx.x;
  int xm = blockIdx.x % 24, c = (blockIdx.x / 24) & 31, b = blockIdx.x / 768;
  float cr = 0.f, ci = 0.f;
  int ib = (b * 32 + c) * 64 * 288 + t;
  for (int x = 0; x < 64; ++x) {
    float br = Bre[ib + x * 288], bi = Bim[ib + x * 288];
    float cx = yc[x * 24 + xm], sx = ys[x * 24 + xm];
    cr += br * cx + bi * sx;
    ci += bi * cx - br * sx;
  }
  int ob = ((b * 32 + c) * 24 + xm) * 288 + t;
  Cre[ob] = cr;
  Cim[ob] = ci;
}

// ------------------- complex channel mix (32->32 per mode) -----------------
// one block per (corner,m1,m2); threads = (o,m3); weights read once, reused
// across the 4 batches.
__global__ void __launch_bounds__(384)
k_cmix(const float* __restrict__ Cre, const float* __restrict__ Cim,
       const float* __restrict__ Wl, float* __restrict__ Dre,
       float* __restrict__ Dim) {
  __shared__ float sxr[1536], sxi[1536];
  int t = threadIdx.x;
  int corner = blockIdx.x / 144;
  int m1 = (blockIdx.x / 12) % 12, m2 = blockIdx.x % 12;
  int xm = m1 + 12 * (corner & 1), ym = m2 + 12 * (corner >> 1);
  {
    int i = t / 12, m3 = t % 12;
#pragma unroll
    for (int bq = 0; bq < 4; ++bq) {
      int src = ((bq * 32 + i) * 24 + xm) * 288 + ym * 12 + m3;
      sxr[bq * 384 + t] = Cre[src];
      sxi[bq * 384 + t] = Cim[src];
    }
  }
  __syncthreads();
  int o = t / 12, m3 = t % 12;
  const float* wr = Wl + corner * 3538944 + o * 1728 + (m1 * 144 + m2 * 12 + m3);
  const float* wi = wr + 1769472;
  float aR[4] = {0, 0, 0, 0}, aI[4] = {0, 0, 0, 0};
  for (int i = 0; i < 32; ++i) {
    float wre = wr[i * 55296], wim = wi[i * 55296];
#pragma unroll
    for (int bq = 0; bq < 4; ++bq) {
      float xr = sxr[bq * 384 + i * 12 + m3];
      float xi = sxi[bq * 384 + i * 12 + m3];
      aR[bq] += xr * wre - xi * wim;
      aI[bq] += xi * wre + xr * wim;
    }
  }
#pragma unroll
  for (int bq = 0; bq < 4; ++bq) {
    int ob = ((bq * 32 + o) * 24 + xm) * 288 + ym * 12 + m3;
    Dre[ob] = aR[bq];
    Dim[ob] = aI[bq];
  }
}

// ------------------- inverse partial DFT along x ---------------------------
__global__ void __launch_bounds__(288)
k_inv_x(const float* __restrict__ Dre, const float* __restrict__ Dim,
        const float* __restrict__ yc, const float* __restrict__ ys,
        float* __restrict__ Ere, float* __restrict__ Eim) {
  int t = threadIdx.x;
  int x = blockIdx.x & 63, c = (blockIdx.x >> 6) & 31, b = blockIdx.x >> 11;
  float er = 0.f, ei = 0.f;
  int ib = (b * 32 + c) * 24 * 288 + t;
  for (int xm = 0; xm < 24; ++xm) {
    float dr = Dre[ib + xm * 288], di = Dim[ib + xm * 288];
    float cx = yc[x * 24 + xm], sx = ys[x * 24 + xm];
    er += dr * cx - di * sx;
    ei += di * cx + dr * sx;
  }
  int ob = ((b * 32 + c) * 64 + x) * 288 + t;
  Ere[ob] = er;
  Eim[ob] = ei;
}

// ------------------- inverse y + inverse rfft z, fused ---------------------
__global__ void __launch_bounds__(384)
k_inv_yz(const float* __restrict__ Ere, const float* __restrict__ Eim,
         const float* __restrict__ zc, const float* __restrict__ zs,
         const float* __restrict__ yc, const float* __restrict__ ys,
         float* __restrict__ x1f) {
  __shared__ float er_[288], ei_[288];
  __shared__ float Fr[768], Fi[768];
  int t = threadIdx.x;
  int x = blockIdx.x & 63, c = (blockIdx.x >> 6) & 31, b = blockIdx.x >> 11;
  int ib = ((b * 32 + c) * 64 + x) * 288;
  if (t < 288) {
    er_[t] = Ere[ib + t];
    ei_[t] = Eim[ib + t];
  }
  __syncthreads();
  for (int idx = t; idx < 768; idx += 384) {  // F[y][k]
    int y = idx / 12, k = idx % 12;
    float fr = 0.f, fi = 0.f;
    for (int ym = 0; ym < 24; ++ym) {
      float er = er_[ym * 12 + k], ei = ei_[ym * 12 + k];
      float cy = yc[y * 24 + ym], sy = ys[y * 24 + ym];
      fr += er * cy - ei * sy;
      fi += ei * cy + er * sy;
    }
    Fr[idx] = fr;
    Fi[idx] = fi;
  }
  __syncthreads();
  float* op = x1f + ((b * 32 + c) * 64 + x) * 2944;
  const float norm = 1.0f / 188416.0f;  // 1/(64*64*46)
  for (int idx = t; idx < 2944; idx += 384) {
    int y = idx / 46, n = idx % 46;
    float v = Fr[y * 12];
#pragma unroll
    for (int k = 1; k < 12; ++k)
      v += 2.0f * (Fr[y * 12 + k] * zc[n * 12 + k] -
                   Fi[y * 12 + k] * zs[n * 12 + k]);
    op[idx] = v * norm;
  }
}

// ------------- pointwise layer: gelu(W2*gelu(W1*x1+b1)+b2 + Ww*x+bw) -------
// WMMA f32 16x16x4. 128-point tile per block, 16 points per wave.
__global__ void __launch_bounds__(256)
k_point(float* __restrict__ field, const float* __restrict__ x1f,
        const float* __restrict__ w1, const float* __restrict__ b1,
        const float* __restrict__ w2, const float* __restrict__ b2,
        const float* __restrict__ ww, const float* __restrict__ wb, int act) {
  __shared__ float sX[128 * 33];
  __shared__ float sX1[128 * 33];  // reused as output staging
  __shared__ float sW[3][32 * 33];
  __shared__ float sB[3][32];
  __shared__ float sH[8][16 * 33];
  int t = threadIdx.x;
  int b = blockIdx.x / 1472;
  int q0 = (blockIdx.x % 1472) * 128;
  int base = b * (CHN * SPAT) + q0;
  __builtin_prefetch(field + base + t, 0, 1);
  __builtin_prefetch(x1f + base + t, 0, 1);
  for (int idx = t; idx < 4096; idx += 256) {
    int c = idx >> 7, p = idx & 127;
    sX[p * 33 + c] = field[base + c * SPAT + p];
    sX1[p * 33 + c] = x1f[base + c * SPAT + p];
  }
  for (int idx = t; idx < 3072; idx += 256) {
    int m = idx >> 10, e = idx & 1023, o = e >> 5, i = e & 31;
    const float* wsrc = (m == 0) ? w1 : (m == 1) ? w2 : ww;
    sW[m][i * 33 + o] = wsrc[e];  // transpose to [i][o]
  }
  if (t < 96) {
    int m = t >> 5, o = t & 31;
    sB[m][o] = ((m == 0) ? b1 : (m == 1) ? b2 : wb)[o];
  }
  __syncthreads();
  int w = t >> 5, lane = t & 31;
  int pb = w * 16, M = lane & 15, half = lane >> 4;
  v8f h[2], s[2];
  gemm16x32(&sX1[pb * 33], sW[0], sB[0], lane, h);  // h1 = W1*x1 + b1
#pragma unroll
  for (int nt = 0; nt < 2; ++nt)
#pragma unroll
    for (int v = 0; v < 8; ++v)
      sH[w][(v + 8 * half) * 33 + nt * 16 + M] = gelu_f(h[nt][v]);
  asm volatile("" ::: "memory");
  gemm16x32(&sH[w][0], sW[1], sB[1], lane, h);      // h2 = W2*gelu(h1) + b2
  gemm16x32(&sX[pb * 33], sW[2], sB[2], lane, s);   // skip = Ww*x + bw
#pragma unroll
  for (int nt = 0; nt < 2; ++nt)
#pragma unroll
    for (int v = 0; v < 8; ++v) {
      float r = h[nt][v] + s[nt][v];
      if (act) r = gelu_f(r);
      sX1[(pb + v + 8 * half) * 33 + nt * 16 + M] = r;
    }
  __syncthreads();
  for (int idx = t; idx < 4096; idx += 256) {
    int c = idx >> 7, p = idx & 127;
    field[base + c * SPAT + p] = sX1[p * 33 + c];
  }
}

// ------------- final MLP: 32 -> 128 (gelu) -> 6, output transpose ----------
__global__ void __launch_bounds__(256)
k_final(const float* __restrict__ field, const float* __restrict__ qw1,
        const float* __restrict__ qb1, const float* __restrict__ qw2,
        const float* __restrict__ qb2, float* __restrict__ out) {
  __shared__ float sX[128 * 33];
  __shared__ float sB1[128];
  __shared__ float sW2[6 * 128];
  __shared__ float sB2[8];
  __shared__ float sH[8][16 * 65];  // per-wave 16 pts x 64 ch (half pass)
  __shared__ float sO[128 * 6];
  int t = threadIdx.x;
  int b = blockIdx.x / 1472;
  int q0 = (blockIdx.x % 1472) * 128;
  int base = b * (CHN * SPAT) + q0;
  for (int idx = t; idx < 4096; idx += 256) {
    int c = idx >> 7, p = idx & 127;
    sX[p * 33 + c] = field[base + c * SPAT + p];
  }
  for (int idx = t; idx < 768; idx += 256) sW2[idx] = qw2[idx];
  if (t < 128) sB1[t] = qb1[t];
  if (t < 6) sB2[t] = qb2[t];
  __syncthreads();
  int w = t >> 5, lane = t & 31;
  int pb = w * 16, M = lane & 15, half = lane >> 4;
  float acc2[3];
#pragma unroll
  for (int r = 0; r < 3; ++r) acc2[r] = 0.f;
#pragma unroll
  for (int hp = 0; hp < 2; ++hp) {  // 64 hidden channels per half pass
#pragma unroll
    for (int nt = 0; nt < 4; ++nt) {
      int obase = hp * 64 + nt * 16;
      float bv = sB1[obase + M];
      v8f c;
#pragma unroll
      for (int v = 0; v < 8; ++v) c[v] = bv;
#pragma unroll
      for (int k0 = 0; k0 < 32; k0 += 4) {
        v2f a, bb;
        a.x = sX[(pb + M) * 33 + k0 + 2 * half];
        a.y = sX[(pb + M) * 33 + k0 + 2 * half + 1];
        // B direct from global qw1[o][i] (16 KB, L2/L0 resident)
        int o = obase + M;
        bb.x = qw1[o * 32 + k0 + 2 * half];
        bb.y = qw1[o * 32 + k0 + 2 * half + 1];
        c = wmma4(a, bb, c);
      }
#pragma unroll
      for (int v = 0; v < 8; ++v)
        sH[w][(v + 8 * half) * 65 + nt * 16 + M] = gelu_f(c[v]);
    }
    asm volatile("" ::: "memory");
#pragma unroll
    for (int r = 0; r < 3; ++r) {  // 16 pts x 6 outs = 96 = 32 lanes x 3
      int idx = lane + 32 * r;
      int pl = idx / 6, o2 = idx % 6;
      float a = 0.f;
      for (int i = 0; i < 64; ++i)
        a += sH[w][pl * 65 + i] * sW2[o2 * 128 + hp * 64 + i];
      acc2[r] += a;
    }
    asm volatile("" ::: "memory");
  }
#pragma unroll
  for (int r = 0; r < 3; ++r) {
    int idx = lane + 32 * r;
    int pl = idx / 6, o2 = idx % 6;
    sO[(pb + pl) * 6 + o2] = acc2[r] + sB2[o2];
  }
  __syncthreads();
  for (int idx = t; idx < 768; idx += 256) {
    int p = idx / 6, o = idx % 6;
    int q = q0 + p;
    int z = q % 46;
    if (z < NZ0) {
      int y = (q / 46) & 63, xx = q / 2944;
      out[(((b * 64 + xx) * 64 + y) * 40 + z) * 6 + o] = sO[idx];
    }
  }
}

// ---------------------------------------------------------------------------
extern "C" void kernel_launch(void* const* d_in, const int* in_sizes, int n_in,
                              void* d_out, int out_size, void* d_ws,
                              size_t ws_size, hipStream_t stream) {
  (void)in_sizes; (void)n_in; (void)out_size; (void)ws_size;
  const float* x      = (const float*)d_in[0];
  const float* p_w    = (const float*)d_in[1];
  const float* p_b    = (const float*)d_in[2];
  const float* conv_w = (const float*)d_in[3];
  const float* mlp_w1 = (const float*)d_in[4];
  const float* mlp_b1 = (const float*)d_in[5];
  const float* mlp_w2 = (const float*)d_in[6];
  const float* mlp_b2 = (const float*)d_in[7];
  const float* w_w    = (const float*)d_in[8];
  const float* w_b    = (const float*)d_in[9];
  const float* q_w1   = (const float*)d_in[10];
  const float* q_b1   = (const float*)d_in[11];
  const float* q_w2   = (const float*)d_in[12];
  const float* q_b2   = (const float*)d_in[13];

  float* ws = (float*)d_ws;  // ~226 MB total
  float* buf0 = ws;                 // current field (B,C,X,Y,Zp)
  float* buf1 = buf0 + FIELD;       // spectral-conv output field
  float* BEre = buf1 + FIELD;       // shared B (fwd-y) / E (inv-x) spectra
  float* BEim = BEre + NBE;
  float* Cre  = BEim + NBE;
  float* Cim  = Cre + NC;
  float* Dre  = Cim + NC;
  float* Dim  = Dre + NC;
  float* zc   = Dim + NC;           // twiddle tables
  float* zs   = zc + 552;
  float* yc   = zs + 552;
  float* ys   = yc + 1536;

  k_tables<<<8, 256, 0, stream>>>(zc, zs, yc, ys);
  k_lift<<<2944, 256, 0, stream>>>(x, p_w, p_b, buf0);
  for (int l = 0; l < 4; ++l) {
    k_fwd_zy<<<8192, 384, 0, stream>>>(buf0, zc, zs, yc, ys, BEre, BEim);
    k_fwd_x<<<3072, 288, 0, stream>>>(BEre, BEim, yc, ys, Cre, Cim);
    k_cmix<<<576, 384, 0, stream>>>(Cre, Cim, conv_w + (size_t)l * 14155776,
                                    Dre, Dim);
    k_inv_x<<<8192, 288, 0, stream>>>(Dre, Dim, yc, ys, BEre, BEim);
    k_inv_yz<<<8192, 384, 0, stream>>>(BEre, BEim, zc, zs, yc, ys, buf1);
    k_point<<<5888, 256, 0, stream>>>(buf0, buf1, mlp_w1 + l * 1024,
                                      mlp_b1 + l * 32, mlp_w2 + l * 1024,
                                      mlp_b2 + l * 32, w_w + l * 1024,
                                      w_b + l * 32, (l < 3) ? 1 : 0);
  }
  k_final<<<5888, 256, 0, stream>>>(buf0, q_w1, q_b1, q_w2, q_b2,
                                    (float*)d_out);
}